// HeteroGNN_25537875542118
// MI455X (gfx1250) — compile-verified
//
#include <hip/hip_runtime.h>

// ---------------------------------------------------------------------------
// HeteroGNN forward for MI455X (gfx1250, wave32, WMMA + async-LDS staging).
//  * Fold Wupd into Wsrc/Wdst (exact linear algebra) -> halves GEMM FLOPs.
//  * GEMMs via V_WMMA_F32_16X16X4_F32 (exact fp32, matches reference).
//  * Big GEMM stages A/B tiles in LDS via GLOBAL_LOAD_ASYNC_TO_LDS_B128
//    (ASYNCcnt DMA path), WMMA fragments fed by conflict-free ds_load_b64.
//  * Scatter-mean via global_atomic_add_f32; BN+LeakyReLU fused elementwise.
// ---------------------------------------------------------------------------

typedef __attribute__((ext_vector_type(2))) float v2f;
typedef __attribute__((ext_vector_type(8))) float v8f;

#define HDIM   256
#define NNODES 50000
#define LRELU  0.01f
#define BNEPS  1e-5f
#define KC     64    // K-chunk staged in LDS
#define PAD    68    // LDS row stride in floats: 272B = 16B-aligned, bank-conflict-free

__device__ __forceinline__ unsigned lds_off(const void* p) {
  // generic pointer to __shared__ = {aperture_hi32, lds_offset32}
  return (unsigned)(unsigned long long)p;
}

__device__ __forceinline__ void async_copy_b128(const void* gptr, unsigned loff) {
  asm volatile("global_load_async_to_lds_b128 %0, %1, off"
               :: "v"(loff), "v"((unsigned long long)gptr)
               : "memory");
}

__device__ __forceinline__ void wait_async0() {
  asm volatile("s_wait_asynccnt 0" ::: "memory");
}

// ---------------- small WMMA GEMM (weight folding) ----------------
// C[M,N] = A[M,K] * B   where B[k][n] = B[k*ldb+n]   (M,N,K multiples of 64)
__global__ __launch_bounds__(256) void wmma_gemm_small(
    const float* __restrict__ A, const float* __restrict__ B,
    float* __restrict__ C, int M, int N, int K, int lda, int ldb, int ldc)
{
  const int lane = threadIdx.x & 31;
  const int wave = threadIdx.x >> 5;
  const int m0 = blockIdx.x * 64 + (wave >> 1) * 16;
  const int n0 = blockIdx.y * 64 + (wave & 1) * 32;
  const int half = lane >> 4;
  const int l15  = lane & 15;
  const float* Arow = A + (long long)(m0 + l15) * lda;
  v8f acc0 = {}, acc1 = {};
  for (int k0 = 0; k0 < K; k0 += 4) {
    const int kk = k0 + 2 * half;
    v2f a; a.x = Arow[kk]; a.y = Arow[kk + 1];
    v2f b0, b1;
    b0.x = B[(long long)kk * ldb + n0 + l15];
    b0.y = B[(long long)(kk + 1) * ldb + n0 + l15];
    b1.x = B[(long long)kk * ldb + n0 + 16 + l15];
    b1.y = B[(long long)(kk + 1) * ldb + n0 + 16 + l15];
    acc0 = __builtin_amdgcn_wmma_f32_16x16x4_f32(false, a, false, b0,
                                                 (short)0, acc0, false, false);
    acc1 = __builtin_amdgcn_wmma_f32_16x16x4_f32(false, a, false, b1,
                                                 (short)0, acc1, false, false);
  }
#pragma unroll
  for (int v = 0; v < 8; v++) {
    const int m = m0 + v + 8 * half;
    C[(long long)m * ldc + n0 + l15]      = acc0[v];
    C[(long long)m * ldc + n0 + 16 + l15] = acc1[v];
  }
}

// ---------------- big WMMA GEMM: C[M,256] = A[M,K] @ B^T (+bias)(+C) --------
// B is [256, K] row-major (folded weights). Block tile 64(M) x 128(N),
// K staged in LDS in KC=64 chunks via async-to-LDS DMA.
// Wave tile: 16(M) x 64(N) = 4 accumulators; 8 waves = 4(M) x 2(N).
__global__ __launch_bounds__(256) void wmma_gemm_big(
    const float* __restrict__ A, const float* __restrict__ B,
    const float* __restrict__ bias, float* __restrict__ C,
    int M, int K, int accumulate)
{
  __shared__ float sA[64 * PAD];
  __shared__ float sB[128 * PAD];
  const int tid  = threadIdx.x;
  const int lane = tid & 31;
  const int wave = tid >> 5;
  const int wm   = wave & 3;   // M sub-tile (16 rows)
  const int wn   = wave >> 2;  // N half (64 cols)
  const int half = lane >> 4;
  const int l15  = lane & 15;
  const int m0 = blockIdx.x * 64;
  const int n0 = blockIdx.y * 128;

  v8f acc[4] = {{}, {}, {}, {}};

  // LDS fragment base pointers (fixed across K chunks)
  const float* pA  = &sA[(wm * 16 + l15) * PAD];
  const float* pB0 = &sB[(wn * 64 +  0 + l15) * PAD];
  const float* pB1 = &sB[(wn * 64 + 16 + l15) * PAD];
  const float* pB2 = &sB[(wn * 64 + 32 + l15) * PAD];
  const float* pB3 = &sB[(wn * 64 + 48 + l15) * PAD];

  for (int kc = 0; kc < K; kc += KC) {
    __syncthreads();  // previous chunk's LDS reads complete before overwrite
    // ---- stage A chunk: 64 rows x 64 floats (row-clamped for M tail) ----
#pragma unroll
    for (int i = 0; i < 4; i++) {
      const int o   = tid + i * 256;      // 0..1023
      const int row = o >> 4;             // 0..63
      const int q   = (o & 15) * 4;       // float offset within chunk
      int r = m0 + row; if (r >= M) r = M - 1;
      async_copy_b128(A + (long long)r * K + kc + q,
                      lds_off(&sA[row * PAD + q]));
    }
    // ---- stage B chunk: 128 rows x 64 floats ----
#pragma unroll
    for (int i = 0; i < 8; i++) {
      const int o   = tid + i * 256;      // 0..2047
      const int row = o >> 4;             // 0..127
      const int q   = (o & 15) * 4;
      async_copy_b128(B + (long long)(n0 + row) * K + kc + q,
                      lds_off(&sB[row * PAD + q]));
    }
    wait_async0();
    __syncthreads();

    // ---- compute on staged chunk ----
#pragma unroll
    for (int k = 0; k < KC; k += 4) {
      const int kk = k + 2 * half;
      const v2f a  = *(const v2f*)(pA  + kk);
      const v2f b0 = *(const v2f*)(pB0 + kk);
      const v2f b1 = *(const v2f*)(pB1 + kk);
      const v2f b2 = *(const v2f*)(pB2 + kk);
      const v2f b3 = *(const v2f*)(pB3 + kk);
      acc[0] = __builtin_amdgcn_wmma_f32_16x16x4_f32(false, a, false, b0,
                                                     (short)0, acc[0], false, false);
      acc[1] = __builtin_amdgcn_wmma_f32_16x16x4_f32(false, a, false, b1,
                                                     (short)0, acc[1], false, false);
      acc[2] = __builtin_amdgcn_wmma_f32_16x16x4_f32(false, a, false, b2,
                                                     (short)0, acc[2], false, false);
      acc[3] = __builtin_amdgcn_wmma_f32_16x16x4_f32(false, a, false, b3,
                                                     (short)0, acc[3], false, false);
    }
  }

  // ---- epilogue: bias, optional accumulate, guarded stores ----
#pragma unroll
  for (int nt = 0; nt < 4; nt++) {
    const int col = n0 + wn * 64 + nt * 16 + l15;
    const float bi = bias ? bias[col] : 0.0f;
#pragma unroll
    for (int v = 0; v < 8; v++) {
      const int m = m0 + wm * 16 + v + 8 * half;
      if (m < M) {
        const long long idx = (long long)m * HDIM + col;
        float r = acc[nt][v] + bi;
        if (accumulate) r += C[idx];
        C[idx] = r;
      }
    }
  }
}

// ---------------- effective bias: beff = WupdL*bdst + WupdR*bsrc + bupd ------
__global__ __launch_bounds__(256) void beff_k(
    const float* __restrict__ Wupd, const float* __restrict__ bdst,
    const float* __restrict__ bsrc, const float* __restrict__ bupd,
    float* __restrict__ beff)
{
  const int h = threadIdx.x;  // blockDim.x == HDIM
  float s = bupd[h];
  const float* row = Wupd + (long long)h * (2 * HDIM);
  for (int m = 0; m < HDIM; m++)
    s += row[m] * bdst[m] + row[HDIM + m] * bsrc[m];
  beff[h] = s;
}

// ---------------- scatter-add (segment_sum) + edge counts --------------------
__global__ __launch_bounds__(256) void scatter_add_k(
    const float* __restrict__ x, const int* __restrict__ edge,
    float* __restrict__ aggr, float* __restrict__ cnt,
    long long total, int E, int shift, int mask)
{
  const long long t = (long long)blockIdx.x * blockDim.x + threadIdx.x;
  if (t >= total) return;
  const int e = (int)(t >> shift);
  const int j = (int)(t & mask);
  const int s = edge[e];
  const int d = edge[E + e];
  atomicAdd(&aggr[((long long)d << shift) + j], x[((long long)s << shift) + j]);
  if (j == 0) atomicAdd(&cnt[d], 1.0f);
}

// ---------------- mean = sum / max(cnt, 1) -----------------------------------
__global__ __launch_bounds__(256) void div_cnt_k(
    float* __restrict__ aggr, const float* __restrict__ cnt,
    long long total, int shift)
{
  const long long t = (long long)blockIdx.x * blockDim.x + threadIdx.x;
  if (t >= total) return;
  const int i = (int)(t >> shift);
  aggr[t] /= fmaxf(cnt[i], 1.0f);
}

// ---------------- BN stats: per-column sum / sumsq ---------------------------
__global__ __launch_bounds__(256) void bn_stats_k(
    const float* __restrict__ e, float* __restrict__ bnsum,
    int Nrows, float prescale)
{
  const int c = threadIdx.x;  // blockDim.x == HDIM
  const int r0 = blockIdx.x * 256;
  const int r1 = (r0 + 256 < Nrows) ? (r0 + 256) : Nrows;
  float s = 0.0f, s2 = 0.0f;
  for (int r = r0; r < r1; r++) {
    const float x = e[(long long)r * HDIM + c] * prescale;
    s += x;
    s2 += x * x;
  }
  atomicAdd(&bnsum[c], s);
  atomicAdd(&bnsum[HDIM + c], s2);
}

// ---------------- fused BN normalize + LeakyReLU -----------------------------
__global__ __launch_bounds__(256) void bn_apply_k(
    const float* __restrict__ e, const float* __restrict__ bnsum,
    const float* __restrict__ g, const float* __restrict__ b,
    float* __restrict__ out, long long total, float prescale, float invN)
{
  const long long t = (long long)blockIdx.x * blockDim.x + threadIdx.x;
  if (t >= total) return;
  const int c = (int)(t & (HDIM - 1));
  const float m  = bnsum[c] * invN;
  const float v  = bnsum[HDIM + c] * invN - m * m;
  const float a  = g[c] * rsqrtf(v + BNEPS);
  const float sh = b[c] - a * m;
  const float x  = e[t] * prescale;
  const float y  = a * x + sh;
  out[t] = (y > 0.0f) ? y : LRELU * y;
}

// ===========================================================================
// Host-side orchestration
// ===========================================================================
struct ConvParams {
  const float *Wsrc, *bsrc, *Wdst, *bdst, *Wupd, *bupd;
  int d_src, d_dst;
};

static void run_conv(const float* x_src, const float* x_dst, const int* edge,
                     int E, const ConvParams& p, float* eout, int accum,
                     float* aggr, float* cnt, float* weffD, float* weffS,
                     float* beff, hipStream_t st)
{
  const int H = HDIM, Nn = NNODES;
  // Weff_dst[H, d_dst] = Wupd[:, :H] @ Wdst ;  Weff_src[H, d_src] = Wupd[:, H:] @ Wsrc
  wmma_gemm_small<<<dim3(H / 64, p.d_dst / 64), 256, 0, st>>>(
      p.Wupd, p.Wdst, weffD, H, p.d_dst, H, 2 * H, p.d_dst, p.d_dst);
  wmma_gemm_small<<<dim3(H / 64, p.d_src / 64), 256, 0, st>>>(
      p.Wupd + H, p.Wsrc, weffS, H, p.d_src, H, 2 * H, p.d_src, p.d_src);
  beff_k<<<1, H, 0, st>>>(p.Wupd, p.bdst, p.bsrc, p.bupd, beff);

  // scatter-mean of x_src over edges into aggr[N, d_src]
  hipMemsetAsync(aggr, 0, (size_t)Nn * p.d_src * sizeof(float), st);
  hipMemsetAsync(cnt, 0, (size_t)Nn * sizeof(float), st);
  const int shift = __builtin_ctz(p.d_src);
  const long long totE = (long long)E << shift;
  scatter_add_k<<<(unsigned)((totE + 255) / 256), 256, 0, st>>>(
      x_src, edge, aggr, cnt, totE, E, shift, p.d_src - 1);
  const long long totN = (long long)Nn << shift;
  div_cnt_k<<<(unsigned)((totN + 255) / 256), 256, 0, st>>>(aggr, cnt, totN, shift);

  // eout = x_dst @ Weff_dst^T + beff  (+eout if accum) ; then += aggr @ Weff_src^T
  const dim3 gbig((Nn + 63) / 64, H / 128);
  wmma_gemm_big<<<gbig, 256, 0, st>>>(x_dst, weffD, beff, eout, Nn, p.d_dst, accum);
  wmma_gemm_big<<<gbig, 256, 0, st>>>(aggr, weffS, nullptr, eout, Nn, p.d_src, 1);
}

static void run_bn_act(const float* e, const float* g, const float* b,
                       float* out, float prescale, float* bnsum, hipStream_t st)
{
  hipMemsetAsync(bnsum, 0, 2 * HDIM * sizeof(float), st);
  bn_stats_k<<<(NNODES + 255) / 256, 256, 0, st>>>(e, bnsum, NNODES, prescale);
  const long long tot = (long long)NNODES * HDIM;
  bn_apply_k<<<(unsigned)((tot + 255) / 256), 256, 0, st>>>(
      e, bnsum, g, b, out, tot, prescale, 1.0f / NNODES);
}

extern "C" void kernel_launch(void* const* d_in, const int* in_sizes, int n_in,
                              void* d_out, int out_size, void* d_ws, size_t ws_size,
                              hipStream_t stream)
{
  (void)n_in; (void)out_size; (void)ws_size;
  const float* x_A = (const float*)d_in[0];
  const float* x_B = (const float*)d_in[1];
  const int* edge_ab = (const int*)d_in[2];
  const int* edge_ba = (const int*)d_in[3];
  const int* edge_aa = (const int*)d_in[4];
  const int E = in_sizes[2] / 2;
  const int H = HDIM, Nn = NNODES, DA = 256, DB = 128;

  auto F = [&](int i) { return (const float*)d_in[i]; };

  // -------- workspace partition (floats); ~257 MB total --------
  float* ws = (float*)d_ws;
  float* aggr  = ws; ws += (size_t)Nn * 256;
  float* cnt   = ws; ws += Nn;
  float* eA    = ws; ws += (size_t)Nn * H;
  float* eB    = ws; ws += (size_t)Nn * H;
  float* hA    = ws; ws += (size_t)Nn * H;
  float* hB    = ws; ws += (size_t)Nn * H;
  float* weffD = ws; ws += (size_t)H * 256;
  float* weffS = ws; ws += (size_t)H * 256;
  float* beff  = ws; ws += H;
  float* bnsum = ws; ws += 2 * H;

  const float* bn_g = F(29);
  const float* bn_b = F(30);

  // ---------------- layer 1 ----------------
  ConvParams ba1{F(11), F(12), F(13), F(14), F(15), F(16), DB, DA};
  ConvParams aa1{F(17), F(18), F(19), F(20), F(21), F(22), DA, DA};
  ConvParams ab1{F(5),  F(6),  F(7),  F(8),  F(9),  F(10), DA, DB};
  run_conv(x_B, x_A, edge_ba, E, ba1, eA, 0, aggr, cnt, weffD, weffS, beff, stream);
  run_conv(x_A, x_A, edge_aa, E, aa1, eA, 1, aggr, cnt, weffD, weffS, beff, stream);
  run_conv(x_A, x_B, edge_ab, E, ab1, eB, 0, aggr, cnt, weffD, weffS, beff, stream);
  run_bn_act(eA, bn_g + 0 * H, bn_b + 0 * H, hA, 0.5f, bnsum, stream);  // (e_ba+e_aa)*0.5
  run_bn_act(eB, bn_g + 1 * H, bn_b + 1 * H, hB, 1.0f, bnsum, stream);

  // ---------------- layer 2 (stacked params: 0=ab, 1=ba, 2=aa) ----------------
  const float* Wsrc2 = F(23); const float* bsrc2 = F(24);
  const float* Wdst2 = F(25); const float* bdst2 = F(26);
  const float* Wupd2 = F(27); const float* bupd2 = F(28);
  auto cp2 = [&](int k) {
    return ConvParams{Wsrc2 + (size_t)k * H * H, bsrc2 + (size_t)k * H,
                      Wdst2 + (size_t)k * H * H, bdst2 + (size_t)k * H,
                      Wupd2 + (size_t)k * H * 2 * H, bupd2 + (size_t)k * H, H, H};
  };
  run_conv(hB, hA, edge_ba, E, cp2(1), eA, 0, aggr, cnt, weffD, weffS, beff, stream);
  run_conv(hA, hA, edge_aa, E, cp2(2), eA, 1, aggr, cnt, weffD, weffS, beff, stream);
  run_conv(hA, hB, edge_ab, E, cp2(0), eB, 0, aggr, cnt, weffD, weffS, beff, stream);

  float* outA = (float*)d_out;
  float* outB = outA + (size_t)Nn * H;
  run_bn_act(eA, bn_g + 2 * H, bn_b + 2 * H, outA, 0.5f, bnsum, stream);
  run_bn_act(eB, bn_g + 3 * H, bn_b + 3 * H, outB, 1.0f, bnsum, stream);
}